// ScatterPillar_2989297238035
// MI455X (gfx1250) — compile-verified
//
#include <hip/hip_runtime.h>
#include <stdint.h>

// Problem constants (match reference)
#define GRID_X 432
#define GRID_Y 496
#define BEV_C  64
#define BATCH  8
#define NPILLARS (BATCH * 12000)          // 96000
#define PLANE  (GRID_Y * GRID_X)          // 214272 floats per (b,c) plane
#define OUT_PER_B ((size_t)BEV_C * PLANE) // 13713408
#define IDX_N  (BATCH * PLANE)            // 1714176 int32 (6.86 MB in d_ws)

// ---------------------------------------------------------------------------
// Kernel 1: initialize the [B, Y, X] pillar-index map to -1 (b128 stores).
// ---------------------------------------------------------------------------
__global__ __launch_bounds__(256) void bev_idx_init(int* __restrict__ idx) {
  int i = blockIdx.x * 256 + threadIdx.x;       // int4 index
  if (i < IDX_N / 4) {                          // IDX_N divisible by 4
    ((int4*)idx)[i] = make_int4(-1, -1, -1, -1);
  }
}

// ---------------------------------------------------------------------------
// Kernel 2: scatter pillar ids into the index map. Random 4B writes, but the
// whole map is 6.9 MB -> L2-resident (192 MB global L2). Coords are unique
// per batch, so no collisions.
//   coords[p] = (cx, cy, cb);  idx[cb][cy][cx] = p
// ---------------------------------------------------------------------------
__global__ __launch_bounds__(256) void bev_idx_scatter(const int* __restrict__ coords,
                                                       int* __restrict__ idx) {
  int p = blockIdx.x * 256 + threadIdx.x;
  if (p >= NPILLARS) return;
  int cx = coords[3 * p + 0];
  int cy = coords[3 * p + 1];
  int cb = coords[3 * p + 2];
  idx[((size_t)cb * GRID_Y + cy) * GRID_X + cx] = p;
}

// ---------------------------------------------------------------------------
// Kernel 3: gather. One block per (b, y). Stage the 432-entry index row into
// LDS with gfx1250 async-to-LDS DMA (108 lanes x b128, tracked by ASYNCcnt),
// then write every output element of out[b][:][y][:] exactly once:
//   occupied -> float4 gather from feats (24.6 MB, L2-resident)
//   empty    -> 0
// Output stores are non-temporal: 439 MB streaming writes must not thrash the
// L2 set holding feats + idx. Stores are lane-contiguous in x -> full-line
// coalesced writes; no read-modify-write anywhere.
// ---------------------------------------------------------------------------
__global__ __launch_bounds__(256) void bev_gather(const float* __restrict__ feats,
                                                  const int* __restrict__ idx,
                                                  float* __restrict__ out) {
  __shared__ int row[GRID_X];                   // 1728 B of the 320 KB WGP LDS
  const int by  = blockIdx.x;                   // 0 .. BATCH*GRID_Y-1
  const int b   = by / GRID_Y;
  const int y   = by - b * GRID_Y;
  const int tid = threadIdx.x;

  // Async DMA the index row into LDS: lanes 0..107, 16 B each.
  const int* grow = idx + (size_t)by * GRID_X;  // uniform -> SADDR sgpr pair
  if (tid < GRID_X / 4) {
    uint32_t lds_addr = (uint32_t)(uintptr_t)&row[4 * tid]; // low 32 bits = DS offset
    uint32_t voff     = 16u * (uint32_t)tid;                // per-lane byte offset
    asm volatile("global_load_async_to_lds_b128 %0, %1, %2"
                 :: "v"(lds_addr), "v"(voff), "s"(grow)
                 : "memory");
  }
  asm volatile("s_wait_asynccnt 0" ::: "memory");
  __syncthreads();

  float*        ob = out + (size_t)b * OUT_PER_B + (size_t)y * GRID_X;
  const float4* f4 = (const float4*)feats;      // feats[p] row = 16 x float4

  for (int x = tid; x < GRID_X; x += 256) {     // 2 iterations (432 / 256)
    const int id = row[x];
    float* o = ob + x;
#pragma unroll 4
    for (int cb4 = 0; cb4 < BEV_C / 4; ++cb4) { // 16 x float4 = all 64 channels
      float4 v = make_float4(0.f, 0.f, 0.f, 0.f);
      if (id >= 0) v = f4[(size_t)id * (BEV_C / 4) + cb4]; // contiguous 16 B gather
      float* oc = o + (size_t)(4 * cb4) * PLANE;
      __builtin_nontemporal_store(v.x, oc);
      __builtin_nontemporal_store(v.y, oc + (size_t)PLANE);
      __builtin_nontemporal_store(v.z, oc + 2 * (size_t)PLANE);
      __builtin_nontemporal_store(v.w, oc + 3 * (size_t)PLANE);
    }
  }
}

// ---------------------------------------------------------------------------
// Host-side launcher. Three dependent kernels on one stream (stream order
// gives the required idx-map ordering). d_ws holds the 6.86 MB index map and
// is fully re-initialized every call (deterministic, graph-capture safe).
// ---------------------------------------------------------------------------
extern "C" void kernel_launch(void* const* d_in, const int* in_sizes, int n_in,
                              void* d_out, int out_size, void* d_ws, size_t ws_size,
                              hipStream_t stream) {
  const float* feats  = (const float*)d_in[0];  // [96000, 64] f32
  const int*   coords = (const int*)d_in[1];    // [96000, 3] int
  float*       out    = (float*)d_out;          // [8, 64, 496, 432] f32
  int*         idx    = (int*)d_ws;             // needs IDX_N * 4 = 6.86 MB

  bev_idx_init   <<<(IDX_N / 4 + 255) / 256, 256, 0, stream>>>(idx);
  bev_idx_scatter<<<(NPILLARS  + 255) / 256, 256, 0, stream>>>(coords, idx);
  bev_gather     <<<BATCH * GRID_Y,          256, 0, stream>>>(feats, idx, out);
}